// MahalanobisCrossEntropyLoss_81209241633406
// MI455X (gfx1250) — compile-verified
//
#include <hip/hip_runtime.h>

// ws layout (bytes):
//   [0..7]    double ce_sum
//   [8..11]   float  pos_sum        (wsf[2])
//   [12..15]  float  s_after        (wsf[3])
//   [16..19]  int    lastneg        (wsi[4])
//   [32..143] float  colsq[28]      (wsf[8..35], padded K 25..27 = 0)
//   [256.. ]  float  norms[P]       (wsf + 64)

typedef __attribute__((ext_vector_type(2))) float v2f;
typedef __attribute__((ext_vector_type(8))) float v8f;

#define TPB 256
#define WAVES_PER_BLK 8
#define TILE_FLOATS 800   // 32 rows * 25 cols
#define TILE_BYTES 3200

__global__ void k_init(const float* __restrict__ M, float* __restrict__ wsf) {
    int t = threadIdx.x;
    if (t < 28) {
        float s = 0.f;
        if (t < 25) {
            for (int i = 0; i < 25; ++i) { float m = M[i * 25 + t]; s += m * m; }
        }
        wsf[8 + t] = s;
    }
    if (t == 28) ((double*)wsf)[0] = 0.0;
    if (t == 29) wsf[2] = 0.f;
    if (t == 30) wsf[3] = 0.f;
    if (t == 31) ((int*)wsf)[4] = -1;
}

// One wave = one 16-pair tile (32 contiguous rows, 3200 contiguous bytes).
// Tile is staged global->LDS with async-to-LDS b64 copies (double buffered,
// ASYNCcnt ordered). CE and the WMMA-A squared-diff operand both read LDS.
__global__ void k_main(const float* __restrict__ X, const int* __restrict__ lab,
                       float* __restrict__ wsf, int P) {
    const int lane   = threadIdx.x & 31;
    const int wv     = threadIdx.x >> 5;
    const int nwaves = (gridDim.x * blockDim.x) >> 5;
    const int gwave  = (blockIdx.x * blockDim.x + threadIdx.x) >> 5;
    const int T      = (P + 15) >> 4;
    const long long Brows = 2ll * (long long)P;

    __shared__ float tile[WAVES_PER_BLK][2][TILE_FLOATS];

    float cs[28];
#pragma unroll
    for (int i = 0; i < 28; ++i) cs[i] = wsf[8 + i];

    // ---- loop-invariant WMMA B operands (colsq in column N=0 only) ----
    // B (4x16): lane 0 holds B[0][0],B[1][0]; lane 16 holds B[2][0],B[3][0].
    // Branch-free via float masks -> pure v_fma, no EXEC manipulation.
    const float m0  = (lane == 0)  ? 1.f : 0.f;
    const float m16 = (lane == 16) ? 1.f : 0.f;
    v2f Bmc[7];
#pragma unroll
    for (int i = 0; i < 7; ++i) {
        const int k0 = 4 * i;
        Bmc[i].x = fmaf(m0, cs[k0],     m16 * cs[k0 + 2]);
        Bmc[i].y = fmaf(m0, cs[k0 + 1], m16 * cs[k0 + 3]);
    }

    float* norms = wsf + 64;

    // clamp for (possible) partial last tile so async copies never run off X
    const unsigned long long xend =
        (unsigned long long)(const char*)X + (unsigned long long)Brows * 100ull - 8ull;

    // issue 13 async b64 global->LDS copies for tile t into buffer buf
    auto issue_copy = [&](int t, int buf) {
        const char* src = (const char*)X + (long long)t * TILE_BYTES;
        unsigned dst = (unsigned)(uintptr_t)&tile[wv][buf][0];
#pragma unroll
        for (int k = 0; k < 12; ++k) {
            unsigned long long g = (unsigned long long)(src + k * 256 + lane * 8);
            if (g > xend) g = xend;
            unsigned l = dst + k * 256 + lane * 8;
            asm volatile("global_load_async_to_lds_b64 %0, %1, off"
                         :: "v"(l), "v"(g) : "memory");
        }
        {   // tail 128 bytes, 16 lanes
            unsigned long long g = (unsigned long long)(src + 3072 + lane * 8);
            if (g > xend) g = xend;
            unsigned l = dst + 3072 + lane * 8;
            if (lane < 16) {
                asm volatile("global_load_async_to_lds_b64 %0, %1, off"
                             :: "v"(l), "v"(g) : "memory");
            }
        }
    };

    double ce_loc  = 0.0;
    float  pos_loc = 0.f;
    int    ln_loc  = -1;

    int t = gwave, buf = 0;
    if (t < T) issue_copy(t, 0);

    for (; t < T; t += nwaves) {
        const int p0 = t << 4;
        const int tn = t + nwaves;
        if (tn < T) {
            issue_copy(tn, buf ^ 1);
            asm volatile("s_wait_asynccnt 0xd" ::: "memory"); // this tile's 13 done
        } else {
            asm volatile("s_wait_asynccnt 0x0" ::: "memory");
        }
        const float* tb = &tile[wv][buf][0];

        // ---------------- cross-entropy: one row per lane (from LDS) --------
        {
            const long long r = 2ll * (long long)p0 + lane;
            if (r < Brows) {
                const float* row = tb + lane * 25;
                float x[25];
#pragma unroll
                for (int c = 0; c < 25; ++c) x[c] = row[c];
                float mx = x[0];
#pragma unroll
                for (int c = 1; c < 25; ++c) mx = fmaxf(mx, x[c]);
                float se = 0.f;
#pragma unroll
                for (int c = 0; c < 25; ++c) se += __expf(x[c] - mx);
                int l = lab[r];
                float xl = row[l];             // single dynamic ds_load
                float lp = xl - mx - __logf(se);
                ce_loc -= (double)lp;
            }
        }

        // ---------------- WMMA norms^2 for the 16-pair tile (from LDS) ------
        // A (32-bit 16x4): lanes 0-15 hold K=k0,k0+1 ; lanes 16-31 hold K=k0+2,k0+3
        const int m   = lane & 15;
        const int off = (lane < 16) ? 0 : 2;
        const float* rA = tb + (2 * m) * 25;
        const float* rB = rA + 25;

        // load all A chunks first so ds_loads clause and WMMAs run back-to-back
        v2f Am[7];
#pragma unroll
        for (int i = 0; i < 6; ++i) {          // chunks 0..5: all columns valid
            const int c0 = 4 * i + off, c1 = c0 + 1;
            float d0 = rA[c0] - rB[c0];
            float d1 = rA[c1] - rB[c1];
            Am[i].x = d0 * d0;
            Am[i].y = d1 * d1;
        }
        {   // chunk k0=24: only K=24 real; select, don't branch
            float d = rA[24] - rB[24];
            Am[6].x = (lane < 16) ? d * d : 0.f;
            Am[6].y = 0.f;
        }

        v8f acc = {};
#pragma unroll
        for (int i = 0; i < 7; ++i) {
            acc = __builtin_amdgcn_wmma_f32_16x16x4_f32(
                false, Am[i], false, Bmc[i], (short)0, acc, false, false);
        }

        // D[:,0]: M=0..7 in lane 0 regs 0..7, M=8..15 in lane 16 regs 0..7.
        float n2 = 0.f;
#pragma unroll
        for (int j = 0; j < 8; ++j) {
            float tv = __shfl(acc[j], (lane < 8) ? 0 : 16, 32);
            n2 = ((lane & 7) == j) ? tv : n2;
        }

        if (lane < 16 && (p0 + lane) < P) {
            int p = p0 + lane;
            float nrm = sqrtf(fmaxf(n2, 0.f));
            norms[p] = nrm;
            int la  = lab[2 * p];
            int lb2 = lab[2 * p + 1];
            if (la == lb2) pos_loc += nrm;
            else           ln_loc = max(ln_loc, p);
        }
        buf ^= 1;
    }

    // ---------------- block reduction, one atomic set per block -------------
    __shared__ double rd[TPB];
    __shared__ float  rf[TPB];
    __shared__ int    ri[TPB];
    int tid = threadIdx.x;
    rd[tid] = ce_loc; rf[tid] = pos_loc; ri[tid] = ln_loc;
    __syncthreads();
    for (int s = TPB / 2; s > 0; s >>= 1) {
        if (tid < s) {
            rd[tid] += rd[tid + s];
            rf[tid] += rf[tid + s];
            ri[tid]  = max(ri[tid], ri[tid + s]);
        }
        __syncthreads();
    }
    if (tid == 0) {
        atomicAdd((double*)wsf, rd[0]);
        atomicAdd(&wsf[2], rf[0]);
        atomicMax(((int*)wsf) + 4, ri[0]);
    }
}

// S_after = sum of norms over same-label pairs with index > lastneg
__global__ void k_after(const int* __restrict__ lab, float* __restrict__ wsf, int P) {
    const int ln = ((const int*)wsf)[4];
    const float* norms = wsf + 64;
    float s_loc = 0.f;
    for (long long p = (long long)blockIdx.x * blockDim.x + threadIdx.x; p < P;
         p += (long long)gridDim.x * blockDim.x) {
        if (p > ln) {
            if (lab[2 * p] == lab[2 * p + 1]) s_loc += norms[p];
        }
    }
    __shared__ float rf[TPB];
    rf[threadIdx.x] = s_loc;
    __syncthreads();
    for (int s = TPB / 2; s > 0; s >>= 1) {
        if (threadIdx.x < s) rf[threadIdx.x] += rf[threadIdx.x + s];
        __syncthreads();
    }
    if (threadIdx.x == 0) atomicAdd(&wsf[3], rf[0]);
}

__global__ void k_final(const float* __restrict__ wsf, float* __restrict__ out, int Btot) {
    double ce = ((const double*)wsf)[0];
    int ln = ((const int*)wsf)[4];
    float metric;
    if (ln >= 0) metric = 1.0f + wsf[3] - wsf[64 + ln];  // 1 + S_after - norms[lastneg]
    else         metric = 1.0f + wsf[2];                 // 1 + distance_p
    metric = fmaxf(metric, 0.f);
    out[0] = (float)(ce / (double)Btot) + 0.01f * metric * 0.5f;
}

extern "C" void kernel_launch(void* const* d_in, const int* in_sizes, int n_in,
                              void* d_out, int out_size, void* d_ws, size_t ws_size,
                              hipStream_t stream) {
    const float* X   = (const float*)d_in[0];
    const int*   lab = (const int*)d_in[1];
    const float* M   = (const float*)d_in[2];
    float* wsf = (float*)d_ws;

    const int Btot = in_sizes[1];   // number of rows (labels count)
    const int P    = Btot / 2;      // number of pairs
    const int T    = (P + 15) / 16; // 16-pair tiles

    k_init<<<1, 64, 0, stream>>>(M, wsf);

    int blocksB = (T + WAVES_PER_BLK - 1) / WAVES_PER_BLK;
    if (blocksB > 2048) blocksB = 2048;
    if (blocksB < 1) blocksB = 1;
    k_main<<<blocksB, TPB, 0, stream>>>(X, lab, wsf, P);

    k_after<<<1024, TPB, 0, stream>>>(lab, wsf, P);

    k_final<<<1, 1, 0, stream>>>(wsf, (float*)d_out, Btot);
}